// QuantumNeuralNetwork_19473381720767
// MI455X (gfx1250) — compile-verified
//
#include <hip/hip_runtime.h>
#include <math.h>

typedef __attribute__((ext_vector_type(2))) float v2f;
typedef __attribute__((ext_vector_type(8))) float v8f;

#define NQ    14
#define NL    6
#define NC    16
#define DIM   16384
#define NBATCH 1024
#define TPB   256

// LDS swizzle: one extra float2 per 16 amplitudes -> stride-16 accesses hit
// distinct banks (lane stride 17*8B = 34 banks).
__device__ __forceinline__ int slot(int i) { return i + (i >> 4); }

__device__ __forceinline__ float2 cmul(float2 a, float2 b) {
  return make_float2(a.x * b.x - a.y * b.y, a.x * b.y + a.y * b.x);
}

// ---------------------------------------------------------------------------
// Kernel 1: precompute the 16x16 complex Kronecker gate matrices.
// 24 blocks = (6 layers) x (4 nibble passes). Pass qubit groups:
//   g=0 (p=0):  q10..q13   g=1 (p=4): q6..q9
//   g=2 (p=8):  q2..q5     g=3 (p=10): q0,q1,I,I
// Layout in d_ws per (l,g): 256 floats Re, then 256 floats Im (row-major 16x16).
// ---------------------------------------------------------------------------
__global__ void qnn_prep(const float* __restrict__ wts, float* __restrict__ Gmat) {
  const int blk = blockIdx.x;          // 0..23
  const int l = blk >> 2, g = blk & 3;
  const int tid = threadIdx.x;         // 0..255 -> one G entry
  const int q0tab[4] = {10, 6, 2, 0};

  float2 U[4][2][2];
#pragma unroll
  for (int f = 0; f < 4; ++f) {
    int qb = (g == 3) ? ((f < 2) ? f : -1) : (q0tab[g] + f);
    if (qb < 0) {
      U[f][0][0] = make_float2(1.f, 0.f); U[f][0][1] = make_float2(0.f, 0.f);
      U[f][1][0] = make_float2(0.f, 0.f); U[f][1][1] = make_float2(1.f, 0.f);
    } else {
      const float* th = wts + (l * NQ + qb) * 3;
      float c0 = cosf(th[0] * 0.5f), s0 = sinf(th[0] * 0.5f);
      float c1 = cosf(th[1] * 0.5f), s1 = sinf(th[1] * 0.5f);
      float c2 = cosf(th[2] * 0.5f), s2 = sinf(th[2] * 0.5f);
      // M = Ry*Rx
      float2 m00 = make_float2( c1 * c0,  s1 * s0);
      float2 m01 = make_float2(-s1 * c0, -c1 * s0);
      float2 m10 = make_float2( s1 * c0, -c1 * s0);
      float2 m11 = make_float2( c1 * c0, -s1 * s0);
      // U = Rz * M
      float2 e0 = make_float2(c2, -s2);
      float2 e1 = make_float2(c2,  s2);
      U[f][0][0] = cmul(e0, m00); U[f][0][1] = cmul(e0, m01);
      U[f][1][0] = cmul(e1, m10); U[f][1][1] = cmul(e1, m11);
    }
  }
  const int r = tid >> 4, c = tid & 15;
  float2 res = make_float2(1.f, 0.f);
#pragma unroll
  for (int f = 0; f < 4; ++f)
    res = cmul(res, U[f][(r >> (3 - f)) & 1][(c >> (3 - f)) & 1]);
  float* Gb = Gmat + (blk << 9);
  Gb[tid]       = res.x;
  Gb[256 + tid] = res.y;
}

// ---------------------------------------------------------------------------
// Kernel 2: one workgroup (256 thr = 8 wave32) per batch element.
// State lives in LDS as interleaved float2 (re,im), swizzled by slot().
// ---------------------------------------------------------------------------
__global__ void __launch_bounds__(TPB) qnn_main(const float* __restrict__ x,
                                                const float* __restrict__ Gmat,
                                                const float* __restrict__ W,
                                                const float* __restrict__ bvec,
                                                float* __restrict__ out) {
  extern __shared__ float2 S[];               // 17408 float2 state (+pad)
  float* red = (float*)(S + 17408);           // 256 floats reduction scratch
  const int tid  = threadIdx.x;
  const int bidx = blockIdx.x;
  const int lane = tid & 31;
  const int wv   = tid >> 5;
  const int half = lane >> 4;                 // which 16-lane half of the wave
  const int rowA = lane & 15;

  // -------- load x, normalize, write state to LDS --------
  const float4* xb = (const float4*)(x + (size_t)bidx * DIM);
  float4 xv[16];
  float ss = 0.f;
#pragma unroll
  for (int i = 0; i < 16; ++i) {
    float4 v = xb[tid + TPB * i];
    xv[i] = v;
    ss += v.x * v.x + v.y * v.y + v.z * v.z + v.w * v.w;
  }
#pragma unroll
  for (int off = 16; off >= 1; off >>= 1) ss += __shfl_xor(ss, off);
  if (lane == 0) red[wv] = ss;
  __syncthreads();
  float tot = 0.f;
#pragma unroll
  for (int w2 = 0; w2 < 8; ++w2) tot += red[w2];
  const float rn = rsqrtf(tot);
#pragma unroll
  for (int i = 0; i < 16; ++i) {
    int j = (tid + TPB * i) << 2;
    S[slot(j + 0)] = make_float2(xv[i].x * rn, 0.f);
    S[slot(j + 1)] = make_float2(xv[i].y * rn, 0.f);
    S[slot(j + 2)] = make_float2(xv[i].z * rn, 0.f);
    S[slot(j + 3)] = make_float2(xv[i].w * rn, 0.f);
  }
  __syncthreads();

  // -------- 6 layers --------
  for (int l = 0; l < NL; ++l) {
    // four fused-4-qubit passes, each: D(16x16) = G(16x16,complex) * S-tile
#pragma unroll
    for (int g = 0; g < 4; ++g) {
      const int p = (g == 0) ? 0 : (g == 1) ? 4 : (g == 2) ? 8 : 10;
      const float2* Gr2 = (const float2*)(Gmat + (((l << 2) + g) << 9));
      const float2* Gi2 = Gr2 + 128;  // +256 floats

      // A operands (ISA 32-bit A 16x4 layout: lanes0-15 K={0,1}, lanes16-31 K={2,3})
      v2f Ar[4], Ai[4], AiN[4];
#pragma unroll
      for (int kc = 0; kc < 4; ++kc) {
        int cidx = (rowA * 16 + 4 * kc + 2 * half) >> 1;
        float2 gr = Gr2[cidx], gi = Gi2[cidx];
        Ar[kc].x  =  gr.x; Ar[kc].y  =  gr.y;
        Ai[kc].x  =  gi.x; Ai[kc].y  =  gi.y;
        AiN[kc].x = -gi.x; AiN[kc].y = -gi.y;   // f32 WMMA has no A/B negate
      }

      const int lowm = (1 << p) - 1;
      // Constant trip count -> wave-uniform loop control, no EXEC masking
      // around the WMMAs (EXEC must be all 1s for WMMA).
      for (int it = 0; it < 8; ++it) {
        const int t = (it << 3) + wv;           // 8 tiles per wave
        int col  = (t << 4) + rowA;             // this lane's column (N)
        int base = (col & lowm) + ((col >> p) << (p + 4));

        // B operands (4x16 rows of the state tile), re/im split
        v2f Br[4], Bi[4];
#pragma unroll
        for (int kc = 0; kc < 4; ++kc) {
          int r0 = 4 * kc + 2 * half;
          float2 e0 = S[slot(base + (r0 << p))];
          float2 e1 = S[slot(base + ((r0 + 1) << p))];
          Br[kc].x = e0.x; Br[kc].y = e1.x;
          Bi[kc].x = e0.y; Bi[kc].y = e1.y;
        }

        // Four independent 4-deep accumulation chains for ILP on the XDL pipe.
        v8f creA = {}, creB = {}, cimA = {}, cimB = {};
#pragma unroll
        for (int kc = 0; kc < 2; ++kc) {
          creA = __builtin_amdgcn_wmma_f32_16x16x4_f32(false, Ar[kc],      false, Br[kc],     (short)0, creA, false, false);
          creA = __builtin_amdgcn_wmma_f32_16x16x4_f32(false, AiN[kc],     false, Bi[kc],     (short)0, creA, false, false);
          creB = __builtin_amdgcn_wmma_f32_16x16x4_f32(false, Ar[kc + 2],  false, Br[kc + 2], (short)0, creB, false, false);
          creB = __builtin_amdgcn_wmma_f32_16x16x4_f32(false, AiN[kc + 2], false, Bi[kc + 2], (short)0, creB, false, false);
          cimA = __builtin_amdgcn_wmma_f32_16x16x4_f32(false, Ar[kc],      false, Bi[kc],     (short)0, cimA, false, false);
          cimA = __builtin_amdgcn_wmma_f32_16x16x4_f32(false, Ai[kc],      false, Br[kc],     (short)0, cimA, false, false);
          cimB = __builtin_amdgcn_wmma_f32_16x16x4_f32(false, Ar[kc + 2],  false, Bi[kc + 2], (short)0, cimB, false, false);
          cimB = __builtin_amdgcn_wmma_f32_16x16x4_f32(false, Ai[kc + 2],  false, Br[kc + 2], (short)0, cimB, false, false);
        }

        // D layout: VGPR v -> M = v + 8*half, N = lane&15
        const int mb = half << 3;
#pragma unroll
        for (int v = 0; v < 8; ++v)
          S[slot(base + ((mb + v) << p))] =
              make_float2(creA[v] + creB[v], cimA[v] + cimB[v]);
      }
      __syncthreads();
    }

    // -------- fused CNOT ring (GF(2)-linear): new[y] = old[x(y)] --------
    // x = (y ^ (y>>1)) with bits 12,13 additionally XOR'ed by y&1.
    float2 stg[64];
#pragma unroll
    for (int k = 0; k < 64; ++k) {
      int y  = (k << 8) | tid;
      int xs = (y ^ (y >> 1)) ^ ((y & 1) ? 0x3000 : 0);
      stg[k] = S[slot(xs)];
    }
    __syncthreads();
#pragma unroll
    for (int k = 0; k < 64; ++k) S[slot((k << 8) | tid)] = stg[k];
    __syncthreads();
  }

  // -------- Pauli-Z expectation values + linear head --------
  float sgn[8];                      // qubits 6..13: sign bit comes from tid
#pragma unroll
  for (int q6 = 0; q6 < 8; ++q6)
    sgn[q6] = ((tid >> (7 - q6)) & 1) ? -1.f : 1.f;

  float acc[14];
#pragma unroll
  for (int q = 0; q < 14; ++q) acc[q] = 0.f;
#pragma unroll
  for (int k = 0; k < 64; ++k) {
    int i = (k << 8) | tid;
    float2 a = S[slot(i)];
    float pr = a.x * a.x + a.y * a.y;
#pragma unroll
    for (int q = 0; q < 6; ++q) {    // qubits 0..5: sign from k (compile-time)
      if ((k >> (5 - q)) & 1) acc[q] -= pr; else acc[q] += pr;
    }
#pragma unroll
    for (int q6 = 0; q6 < 8; ++q6) acc[6 + q6] = fmaf(pr, sgn[q6], acc[6 + q6]);
  }
#pragma unroll
  for (int q = 0; q < 14; ++q) {
    float v = acc[q];
#pragma unroll
    for (int off = 16; off >= 1; off >>= 1) v += __shfl_xor(v, off);
    if (lane == 0) red[wv * 16 + q] = v;
  }
  __syncthreads();

  if (tid < NC) {
    float o = bvec[tid];
#pragma unroll
    for (int q = 0; q < 14; ++q) {
      float ev = 0.f;
#pragma unroll
      for (int w2 = 0; w2 < 8; ++w2) ev += red[w2 * 16 + q];
      o = fmaf(ev, W[tid * 14 + q], o);
    }
    out[(size_t)bidx * NC + tid] = o;
  }
}

// ---------------------------------------------------------------------------
extern "C" void kernel_launch(void* const* d_in, const int* in_sizes, int n_in,
                              void* d_out, int out_size, void* d_ws, size_t ws_size,
                              hipStream_t stream) {
  (void)in_sizes; (void)n_in; (void)out_size; (void)ws_size;
  const float* x   = (const float*)d_in[0];
  const float* wts = (const float*)d_in[1];
  const float* W   = (const float*)d_in[2];
  const float* b   = (const float*)d_in[3];
  float* Gmat = (float*)d_ws;                 // 24 * 512 floats = 48 KB
  float* out  = (float*)d_out;

  const size_t lds_bytes = 17408 * sizeof(float2) + 256 * sizeof(float); // 140288
  (void)hipFuncSetAttribute((const void*)qnn_main,
                            hipFuncAttributeMaxDynamicSharedMemorySize,
                            (int)lds_bytes);

  qnn_prep<<<24, TPB, 0, stream>>>(wts, Gmat);
  qnn_main<<<NBATCH, TPB, lds_bytes, stream>>>(x, Gmat, W, b, out);
}